// CombinedModel_59313498357800
// MI455X (gfx1250) — compile-verified
//
#include <hip/hip_runtime.h>
#include <hip/hip_bf16.h>
#include <math.h>

// ---------------------------------------------------------------------------
// Problem constants (match reference)
// ---------------------------------------------------------------------------
#define N_ROIS   1024
#define BANDS    7
#define M_TOTAL  (N_ROIS * BANDS)      // 7168
#define ROI_H    112
#define ROI_W    224
#define BAND_H   16
#define POOL_H   8
#define POOL_W   16
#define FLAT_DIM 512
#define HID      128
#define HEADS    8
#define PROV     38
#define ALPHA    25
#define AD       35
#define CONF_THRES 0.25f

// d_out layout (flat, reference tuple order): prov, alpha, ad, mask, total
#define OFF_PROV  ((size_t)0)
#define OFF_ALPHA ((size_t)M_TOTAL * PROV)                       // 272384
#define OFF_AD    (OFF_ALPHA + (size_t)M_TOTAL * ALPHA)          // 451584
#define OFF_MASK  (OFF_AD + (size_t)6 * M_TOTAL * AD)            // 1956864
#define OFF_TOTAL (OFF_MASK + (size_t)M_TOTAL)                   // 1964032

typedef __attribute__((ext_vector_type(2))) float v2f;
typedef __attribute__((ext_vector_type(8))) float v8f;

// ---------------------------------------------------------------------------
// Kernel A: per-band crop + adaptive max-pool (16x224 -> 8x16) + conv3x3(3->16)
//           + ReLU + 2x2 maxpool -> flat[m][512]
// One block per band. Image (12MB) is L2-resident; this is bandwidth-trivial.
// ---------------------------------------------------------------------------
__global__ __launch_bounds__(256) void band_pool_conv_kernel(
    const float* __restrict__ x,        // (3,1024,1024)
    const int*   __restrict__ rois,     // (1024,4)
    const float* __restrict__ conv_w,   // (16,3,3,3)
    const float* __restrict__ conv_b,   // (16,)
    float*       __restrict__ flat)     // (M,512)
{
    __shared__ float sP[3 * POOL_H * POOL_W];   // pooled 3x8x16
    __shared__ float sW[16 * 27];
    __shared__ float sB[16];

    const int m    = blockIdx.x;
    const int roi  = m / BANDS;
    const int band = m % BANDS;
    const int t    = threadIdx.x;

    const int x1   = rois[roi * 4 + 0];
    const int y1   = rois[roi * 4 + 1];
    const int row0 = y1 + band * BAND_H;

    for (int i = t; i < 16 * 27; i += 256) sW[i] = conv_w[i];
    if (t < 16) sB[t] = conv_b[t];

    // adaptive max pool: window 2 (rows) x 14 (cols)
    for (int idx = t; idx < 3 * POOL_H * POOL_W; idx += 256) {
        const int c   = idx >> 7;            // /128
        const int rem = idx & 127;
        const int py  = rem >> 4;
        const int px  = rem & 15;
        const float* base = x + (size_t)c * 1024 * 1024
                              + (size_t)(row0 + py * 2) * 1024
                              + (size_t)(x1 + px * 14);
        float mx = -INFINITY;
        #pragma unroll
        for (int r = 0; r < 2; ++r)
            for (int j = 0; j < 14; ++j)
                mx = fmaxf(mx, base[r * 1024 + j]);
        sP[idx] = mx;
    }
    __syncthreads();

    // conv SAME 3x3 on 8x16, relu, 2x2 maxpool -> (16,4,8) = 512 flat values
    for (int f = t; f < FLAT_DIM; f += 256) {
        const int ch = f >> 5;               // 0..15
        const int y4 = (f >> 3) & 3;         // 0..3
        const int x8 = f & 7;                // 0..7
        const float* wch = sW + ch * 27;
        const float bch  = sB[ch];
        float mx = -INFINITY;
        #pragma unroll
        for (int dy = 0; dy < 2; ++dy) {
            #pragma unroll
            for (int dx = 0; dx < 2; ++dx) {
                const int yy = 2 * y4 + dy;
                const int xx = 2 * x8 + dx;
                float acc = bch;
                #pragma unroll
                for (int i = 0; i < 3; ++i)
                    #pragma unroll
                    for (int ky = 0; ky < 3; ++ky)
                        #pragma unroll
                        for (int kx = 0; kx < 3; ++kx) {
                            const int iy = yy + ky - 1;
                            const int ix = xx + kx - 1;
                            if (iy >= 0 && iy < POOL_H && ix >= 0 && ix < POOL_W)
                                acc += sP[i * 128 + iy * 16 + ix] * wch[i * 9 + ky * 3 + kx];
                        }
                mx = fmaxf(mx, fmaxf(acc, 0.0f));
            }
        }
        flat[(size_t)m * FLAT_DIM + f] = mx;
    }
}

// ---------------------------------------------------------------------------
// Kernel B: FC1 GEMM via V_WMMA_F32_16X16X4_F32 (fp32 in, fp32 acc).
//   D(8,M,128) = ReLU(flat(M,512) x w1[head](512,128) + b1[head])
// One wave computes one 16(M) x 16(N) tile, accumulating K=512 in steps of 4
// (128 chained WMMAs). 8 waves per block; EXEC all-ones per wave as required.
//
// Fragment layouts per ISA 7.12.2 (32-bit matrices), lane L, half = L/16:
//   A 16x4 : lane holds A[L%16][k0+2*half], A[L%16][k0+2*half+1]  (float2 load)
//   B 4x16 : lane holds B[k0+2*half][L%16], B[k0+2*half+1][L%16]
//   C 16x16: VGPR r -> M = r + 8*half, N = L%16
// ---------------------------------------------------------------------------
__global__ __launch_bounds__(256) void fc1_wmma_kernel(
    const float* __restrict__ flat,   // (M,512)
    const float* __restrict__ w1,     // (8,512,128)
    const float* __restrict__ b1,     // (8,128)
    float*       __restrict__ h)      // (8,M,128)
{
    const int wave  = threadIdx.x >> 5;            // 0..7
    const int lane  = threadIdx.x & 31;
    const int half  = lane >> 4;                   // 0/1
    const int l16   = lane & 15;

    const int mtile = blockIdx.x;                  // 0..447
    const int ntile = blockIdx.y * 8 + wave;       // 0..63
    const int head  = ntile >> 3;                  // 0..7
    const int ncol  = ((ntile & 7) << 4) + l16;    // 0..127 within head

    const float* Arow = flat + (size_t)(mtile * 16 + l16) * FLAT_DIM + 2 * half;
    const float* Bcol = w1 + (size_t)head * FLAT_DIM * HID
                           + (size_t)(2 * half) * HID + ncol;

    v8f c = {};
    #pragma unroll 8
    for (int k0 = 0; k0 < FLAT_DIM; k0 += 4) {
        v2f a = *(const v2f*)(Arow + k0);          // A[m][k0+2h], A[m][k0+2h+1]
        v2f b;
        b.x = Bcol[(size_t)k0 * HID];              // B[k0+2h][n]
        b.y = Bcol[(size_t)k0 * HID + HID];        // B[k0+2h+1][n]
        c = __builtin_amdgcn_wmma_f32_16x16x4_f32(
                /*neg_a=*/false, a, /*neg_b=*/false, b,
                /*c_mod=*/(short)0, c, /*reuse_a=*/false, /*reuse_b=*/false);
    }

    const float bias = b1[head * HID + ncol];
    #pragma unroll
    for (int r = 0; r < 8; ++r) {
        const int mrow = mtile * 16 + r + 8 * half;
        float v = c[r] + bias;
        v = v > 0.0f ? v : 0.0f;
        h[((size_t)head * M_TOTAL + mrow) * HID + ncol] = v;
    }
}

// ---------------------------------------------------------------------------
// Kernel C: FC2 heads. One block per row m; 8x128 h-rows staged in LDS.
//   t in [0,38)    : prov  = h[0] @ w2_prov  + b2_prov
//   t in [38,63)   : alpha = h[1] @ w2_alpha + b2_alpha
//   t in [63,273)  : ad[j] = h[2+j] @ w2_ad[j] + b2_ad[j],  j=(t-63)/35
// ---------------------------------------------------------------------------
__global__ __launch_bounds__(288) void fc2_kernel(
    const float* __restrict__ h,
    const float* __restrict__ w2_prov,  const float* __restrict__ b2_prov,
    const float* __restrict__ w2_alpha, const float* __restrict__ b2_alpha,
    const float* __restrict__ w2_ad,    const float* __restrict__ b2_ad,
    float* __restrict__ out)
{
    __shared__ float sh[HEADS * HID];
    const int m = blockIdx.x;
    const int t = threadIdx.x;

    for (int i = t; i < HEADS * HID; i += 288) {
        const int k = i >> 7, j = i & 127;
        sh[i] = h[((size_t)k * M_TOTAL + m) * HID + j];
    }
    __syncthreads();

    if (t < PROV) {
        float acc = b2_prov[t];
        #pragma unroll 4
        for (int d = 0; d < HID; ++d) acc += sh[d] * w2_prov[d * PROV + t];
        out[OFF_PROV + (size_t)m * PROV + t] = acc;
    } else if (t < PROV + ALPHA) {
        const int c = t - PROV;
        float acc = b2_alpha[c];
        #pragma unroll 4
        for (int d = 0; d < HID; ++d) acc += sh[HID + d] * w2_alpha[d * ALPHA + c];
        out[OFF_ALPHA + (size_t)m * ALPHA + c] = acc;
    } else if (t < PROV + ALPHA + 6 * AD) {
        const int u = t - (PROV + ALPHA);
        const int j = u / AD;
        const int c = u % AD;
        float acc = b2_ad[j * AD + c];
        const float* hr = sh + (2 + j) * HID;
        const float* wj = w2_ad + (size_t)j * HID * AD;
        #pragma unroll 4
        for (int d = 0; d < HID; ++d) acc += hr[d] * wj[d * AD + c];
        out[OFF_AD + ((size_t)j * M_TOTAL + m) * AD + c] = acc;
    }
}

// ---------------------------------------------------------------------------
// Kernel D/E: losses. Per-row cross entropies, mask, masked sum -> total.
// ---------------------------------------------------------------------------
__device__ __forceinline__ float ce_row(const float* __restrict__ logits,
                                        int n, int tgt)
{
    float mx = logits[0];
    for (int i = 1; i < n; ++i) mx = fmaxf(mx, logits[i]);
    float s = 0.0f;
    for (int i = 0; i < n; ++i) s += __expf(logits[i] - mx);
    return (mx + __logf(s)) - logits[tgt];
}

__global__ void zero_total_kernel(float* __restrict__ total)
{
    if (blockIdx.x == 0 && threadIdx.x == 0) *total = 0.0f;
}

__global__ __launch_bounds__(256) void loss_kernel(
    const float* __restrict__ out,
    const int*   __restrict__ det_t,   // (1024,)
    const int*   __restrict__ cls_t,   // (1024,8)
    float* __restrict__ mask_out,
    float* __restrict__ total)
{
    const int m = blockIdx.x * blockDim.x + threadIdx.x;
    if (m >= M_TOTAL) return;
    const int roi = m / BANDS;

    const float* prov  = out + OFF_PROV  + (size_t)m * PROV;
    const float* alpha = out + OFF_ALPHA + (size_t)m * ALPHA;

    const float det_loss = ce_row(prov, PROV, det_t[roi]);

    float cls = ce_row(prov, PROV, cls_t[roi * 8 + 0]);
    cls += ce_row(alpha, ALPHA, cls_t[roi * 8 + 1]);
    #pragma unroll
    for (int j = 0; j < 6; ++j) {
        const float* ad = out + OFF_AD + ((size_t)j * M_TOTAL + m) * AD;
        cls += ce_row(ad, AD, cls_t[roi * 8 + 2 + j]);
    }

    const bool msk = det_loss < CONF_THRES;
    mask_out[m] = msk ? 1.0f : 0.0f;
    const float contrib = msk ? cls : 0.0f;
    atomicAdd(total, contrib);
}

// ---------------------------------------------------------------------------
// Launch
// ---------------------------------------------------------------------------
extern "C" void kernel_launch(void* const* d_in, const int* in_sizes, int n_in,
                              void* d_out, int out_size, void* d_ws, size_t ws_size,
                              hipStream_t stream)
{
    (void)in_sizes; (void)n_in; (void)out_size; (void)ws_size;

    const float* x        = (const float*)d_in[0];
    const int*   rois     = (const int*)  d_in[1];
    const int*   det_t    = (const int*)  d_in[2];
    const int*   cls_t    = (const int*)  d_in[3];
    const float* conv_w   = (const float*)d_in[4];
    const float* conv_b   = (const float*)d_in[5];
    const float* w1       = (const float*)d_in[6];
    const float* b1       = (const float*)d_in[7];
    const float* w2_prov  = (const float*)d_in[8];
    const float* b2_prov  = (const float*)d_in[9];
    const float* w2_alpha = (const float*)d_in[10];
    const float* b2_alpha = (const float*)d_in[11];
    const float* w2_ad    = (const float*)d_in[12];
    const float* b2_ad    = (const float*)d_in[13];

    float* out  = (float*)d_out;
    float* flat = (float*)d_ws;                                  // M*512
    float* h    = flat + (size_t)M_TOTAL * FLAT_DIM;             // 8*M*128

    band_pool_conv_kernel<<<M_TOTAL, 256, 0, stream>>>(x, rois, conv_w, conv_b, flat);

    dim3 g1(M_TOTAL / 16, 64 / 8);   // 448 M-tiles x 8 blocks (8 waves each = 64 N-tiles)
    fc1_wmma_kernel<<<g1, 256, 0, stream>>>(flat, w1, b1, h);

    fc2_kernel<<<M_TOTAL, 288, 0, stream>>>(h, w2_prov, b2_prov,
                                            w2_alpha, b2_alpha,
                                            w2_ad, b2_ad, out);

    zero_total_kernel<<<1, 1, 0, stream>>>(out + OFF_TOTAL);
    loss_kernel<<<(M_TOTAL + 255) / 256, 256, 0, stream>>>(out, det_t, cls_t,
                                                           out + OFF_MASK,
                                                           out + OFF_TOTAL);
}